// GraphActorCritic_21835613732995
// MI455X (gfx1250) — compile-verified
//
#include <hip/hip_runtime.h>
#include <hip/hip_bf16.h>

typedef __attribute__((ext_vector_type(2))) float v2f;
typedef __attribute__((ext_vector_type(8))) float v8f;

#define NN 100000
#define NE 800000
#define HD 128
#define FIN 9

// ---------------------------------------------------------------- reductions
__global__ void k_reduce_est(const float* __restrict__ est, float* __restrict__ sums, int n) {
    __shared__ float s0[256], s1[256];
    int i = blockIdx.x * 256 + threadIdx.x;
    float v = (i < n) ? est[i] : 0.f;
    s0[threadIdx.x] = v;
    s1[threadIdx.x] = v * v;
    __syncthreads();
    for (int off = 128; off > 0; off >>= 1) {
        if (threadIdx.x < off) {
            s0[threadIdx.x] += s0[threadIdx.x + off];
            s1[threadIdx.x] += s1[threadIdx.x + off];
        }
        __syncthreads();
    }
    if (threadIdx.x == 0) {
        atomicAdd(&sums[0], s0[0]);
        atomicAdd(&sums[1], s1[0]);
    }
}

// --------------------------------------------------------- edge degree/stats
__global__ void k_edge_stats(const int* __restrict__ edges, const int* __restrict__ status,
                             float* __restrict__ in_deg, float* __restrict__ out_deg,
                             float* __restrict__ in_stats, float* __restrict__ out_stats, int e) {
    int i = blockIdx.x * blockDim.x + threadIdx.x;
    if (i >= e) return;
    int s = edges[2 * i], d = edges[2 * i + 1];
    atomicAdd(&out_deg[s], 1.f);
    atomicAdd(&in_deg[d], 1.f);
    atomicAdd(&in_stats[d * 3 + status[s]], 1.f);
    atomicAdd(&out_stats[s * 3 + status[d]], 1.f);
}

// --------------------------------------------------------------- node feats
__global__ void k_node_feats(const int* __restrict__ status_raw, const float* __restrict__ est,
                             const float* __restrict__ in_deg, const float* __restrict__ out_deg,
                             const float* __restrict__ in_stats, const float* __restrict__ out_stats,
                             const float* __restrict__ sums, float* __restrict__ feats,
                             float* __restrict__ dinv, int* __restrict__ n1, int n) {
    int i = blockIdx.x * blockDim.x + threadIdx.x;
    if (i >= n) return;
    float sum = sums[0], sumsq = sums[1];
    float mean = sum / (float)n;
    float var = (sumsq - sum * mean) / (float)(n - 1);   // ddof=1
    var = fmaxf(var, 0.f);
    float sd = sqrtf(var);
    float denom = (sd > 1e-5f) ? sd : 1.f;
    int raw = status_raw[i];
    float* f = feats + (size_t)i * FIN;
    f[0] = (float)(raw - 1);
    f[1] = (est[i] - mean) / denom;
    f[2] = in_deg[i] + out_deg[i];
    f[3] = in_stats[3 * i + 0];
    f[4] = in_stats[3 * i + 1];
    f[5] = in_stats[3 * i + 2];
    f[6] = out_stats[3 * i + 0];
    f[7] = out_stats[3 * i + 1];
    f[8] = out_stats[3 * i + 2];
    dinv[i] = rsqrtf(in_deg[i] + 1.f);   // deg includes self-loop
    if (raw == 2) atomicAdd(n1, 1);
}

// ----------------------------------------------------- feats @ W1 (K=9, VALU)
__global__ void k_gemm_fin(const float* __restrict__ feats, const float* __restrict__ W1,
                           float* __restrict__ out, int n) {
    int idx = blockIdx.x * blockDim.x + threadIdx.x;
    if (idx >= n * HD) return;
    int i = idx >> 7, j = idx & 127;
    const float* f = feats + (size_t)i * FIN;
    float acc = 0.f;
#pragma unroll
    for (int k = 0; k < FIN; k++) acc = fmaf(f[k], W1[k * HD + j], acc);
    out[idx] = acc;
}

// -------------------------------------------------- GCN scatter: self + edges
__global__ void k_selfloop(const float* __restrict__ h, const float* __restrict__ dinv,
                           float* __restrict__ agg, int n) {
    int idx = blockIdx.x * blockDim.x + threadIdx.x;
    if (idx >= n * HD) return;
    float di = dinv[idx >> 7];
    agg[idx] = h[idx] * di * di;
}

__global__ void k_edge_agg(const int* __restrict__ edges, const float* __restrict__ h,
                           const float* __restrict__ dinv, float* __restrict__ agg, int e) {
    int idx = blockIdx.x * blockDim.x + threadIdx.x;
    int ei = idx >> 7;
    if (ei >= e) return;
    int f = idx & 127;
    int s = edges[2 * ei], d = edges[2 * ei + 1];
    float nrm = dinv[s] * dinv[d];
    atomicAdd(&agg[(size_t)d * HD + f], h[(size_t)s * HD + f] * nrm);
}

__global__ void k_bias_relu(float* __restrict__ x, const float* __restrict__ b, int n) {
    int idx = blockIdx.x * blockDim.x + threadIdx.x;
    if (idx >= n * HD) return;
    x[idx] = fmaxf(x[idx] + b[idx & 127], 0.f);
}

// ------------------------------------------- N x 128 @ 128 x 128 via f32 WMMA
// Each wave owns one 16x16 tile of Y; block = 8 waves = 16 rows x 128 cols.
// K looped in steps of 4 via V_WMMA_F32_16X16X4_F32 (fp32 accuracy preserved).
// W staged in LDS (64 KB); grid-stride over 16-row chunks amortizes the fill.
// n must be a multiple of 16 (100000 = 6250*16): EXEC stays all-ones in loop.
__global__ void k_gemm_wmma(const float* __restrict__ X, const float* __restrict__ W,
                            const float* __restrict__ addv, int do_relu,
                            float* __restrict__ Y, int n) {
    extern __shared__ float sW[];                    // HD*HD floats
    for (int t = threadIdx.x; t < HD * HD; t += blockDim.x) sW[t] = W[t];
    __syncthreads();

    int wave = threadIdx.x >> 5;
    int lane = threadIdx.x & 31;
    int n0 = wave * 16;           // column tile
    int lhalf = lane >> 4;        // 0 | 1
    int l15 = lane & 15;
    int col = n0 + l15;
    float av = addv ? addv[col] : 0.f;

    for (int row0 = blockIdx.x * 16; row0 < n; row0 += gridDim.x * 16) {
        v8f acc = {0.f, 0.f, 0.f, 0.f, 0.f, 0.f, 0.f, 0.f};
        const float* xrow = X + (size_t)(row0 + l15) * HD;
#pragma unroll 8
        for (int k = 0; k < HD; k += 4) {
            int ka = k + 2 * lhalf;                  // A: lanes 0-15 -> K=k,k+1 ; 16-31 -> k+2,k+3
            v2f a, b;
            a.x = xrow[ka];
            a.y = xrow[ka + 1];
            b.x = sW[ka * HD + col];                 // B: row-per-VGPR striped across lane halves
            b.y = sW[(ka + 1) * HD + col];
            acc = __builtin_amdgcn_wmma_f32_16x16x4_f32(
                false, a, false, b, (short)0, acc, false, false);
        }
#pragma unroll
        for (int r = 0; r < 8; r++) {                // C: VGPR r -> rows r / r+8 per lane half
            int row = row0 + r + 8 * lhalf;
            float v = acc[r] + av;
            if (do_relu) v = fmaxf(v, 0.f);
            Y[(size_t)row * HD + col] = v;
        }
    }
}

// ------------------------------------------------------------ column mean(s)
__global__ void k_colsum(const float* __restrict__ x, float* __restrict__ gsum, int n) {
    int j = threadIdx.x;                              // 128 threads
    int r0 = blockIdx.x * 512;
    int r1 = r0 + 512; if (r1 > n) r1 = n;
    float acc = 0.f;
    for (int r = r0; r < r1; r++) acc += x[(size_t)r * HD + j];
    atomicAdd(&gsum[j], acc);
}

// g = gsum/N ; gterm = g @ Wa1[128:256,:] + ba1   (constant part of actor MLP)
__global__ void k_gterm(const float* __restrict__ gsum, const float* __restrict__ Wa1,
                        const float* __restrict__ ba1, float* __restrict__ gmean,
                        float* __restrict__ gterm, int n) {
    __shared__ float sg[HD];
    int j = threadIdx.x;                              // 128 threads, 1 block
    sg[j] = gsum[j] / (float)n;
    __syncthreads();
    float acc = ba1[j];
    for (int k = 0; k < HD; k++) acc = fmaf(sg[k], Wa1[(HD + k) * HD + j], acc);
    gmean[j] = sg[j];
    gterm[j] = acc;
}

// ---------------------------------------------- actor: dot(hA, Wa2) + masking
__global__ void k_logits(const float* __restrict__ hA, const float* __restrict__ Wa2,
                         const float* __restrict__ ba2, const int* __restrict__ status_raw,
                         const int* __restrict__ n1, float* __restrict__ logits, int n) {
    int wave = threadIdx.x >> 5, lane = threadIdx.x & 31;
    int node = blockIdx.x * 8 + wave;
    if (node >= n) return;
    const float* hr = hA + (size_t)node * HD;
    float acc = 0.f;
    for (int k = lane; k < HD; k += 32) acc = fmaf(hr[k], Wa2[k], acc);
    for (int off = 16; off > 0; off >>= 1) acc += __shfl_xor(acc, off, 32);
    if (lane == 0) {
        float logit = acc + ba2[0];
        int raw = status_raw[node];
        float mask = (*n1 > 0) ? (raw == 2 ? 1.f : 0.f) : (raw == 1 ? 1.f : 0.f);
        logits[node] = logit * mask + (1.f - mask) * -1e9f;
    }
}

// ------------------------------------------------------- value head (1 block)
__global__ void k_value(const float* __restrict__ gmean, const float* __restrict__ Wc1,
                        const float* __restrict__ bc1, const float* __restrict__ Wc2,
                        const float* __restrict__ bc2, const float* __restrict__ Wc3,
                        const float* __restrict__ bc3, float* __restrict__ out_value) {
    __shared__ float pooled[256], h1[256], h2[128];
    int t = threadIdx.x;                              // 256 threads
    pooled[t] = gmean[t & 127];                       // pooled = [g, g]
    __syncthreads();
    float acc = bc1[t];
    for (int k = 0; k < 256; k++) acc = fmaf(pooled[k], Wc1[k * 256 + t], acc);
    h1[t] = fmaxf(acc, 0.f);
    __syncthreads();
    if (t < 128) {
        float a2 = bc2[t];
        for (int k = 0; k < 256; k++) a2 = fmaf(h1[k], Wc2[k * 128 + t], a2);
        h2[t] = fmaxf(a2, 0.f);
    }
    __syncthreads();
    if (t == 0) {
        float v = bc3[0];
        for (int k = 0; k < 128; k++) v = fmaf(h2[k], Wc3[k], v);
        *out_value = v;
    }
}

// ===========================================================================
extern "C" void kernel_launch(void* const* d_in, const int* in_sizes, int n_in,
                              void* d_out, int out_size, void* d_ws, size_t ws_size,
                              hipStream_t stream) {
    const int*   status = (const int*)  d_in[0];
    const float* est    = (const float*)d_in[1];
    const int*   edges  = (const int*)  d_in[2];
    const float* W1  = (const float*)d_in[3];
    const float* b1  = (const float*)d_in[4];
    const float* W2  = (const float*)d_in[5];
    const float* b2  = (const float*)d_in[6];
    const float* Wa1 = (const float*)d_in[7];
    const float* ba1 = (const float*)d_in[8];
    const float* Wa2 = (const float*)d_in[9];
    const float* ba2 = (const float*)d_in[10];
    const float* Wc1 = (const float*)d_in[11];
    const float* bc1 = (const float*)d_in[12];
    const float* Wc2 = (const float*)d_in[13];
    const float* bc2 = (const float*)d_in[14];
    const float* Wc3 = (const float*)d_in[15];
    const float* bc3 = (const float*)d_in[16];

    const int n = in_sizes[0];        // NN
    const int e = in_sizes[2] / 2;    // NE

    // ---- workspace layout
    char* ws = (char*)d_ws;
    float* sums   = (float*)(ws + 0);        // [sum, sumsq]
    int*   n1     = (int*)  (ws + 8);
    float* gsum   = (float*)(ws + 256);      // 128
    float* gmean  = (float*)(ws + 1024);     // 128
    float* gterm  = (float*)(ws + 2048);     // 128
    size_t o = 4096;
    float* in_deg    = (float*)(ws + o); o += (size_t)n * 4;
    float* out_deg   = (float*)(ws + o); o += (size_t)n * 4;
    float* in_stats  = (float*)(ws + o); o += (size_t)n * 12;
    float* out_stats = (float*)(ws + o); o += (size_t)n * 12;
    size_t zero_bytes = o - 4096;            // 8N floats, contiguous
    float* dinv  = (float*)(ws + o); o += (size_t)n * 4;
    float* feats = (float*)(ws + o); o += (size_t)n * FIN * 4;
    float* bufA  = (float*)(ws + o); o += (size_t)n * HD * 4;
    float* bufB  = (float*)(ws + o); o += (size_t)n * HD * 4;

    float* logits = (float*)d_out;
    float* value  = (float*)d_out + n;

    // ---- zero accumulators (graph-capture-safe, stream-ordered)
    hipMemsetAsync(ws, 0, 4096, stream);
    hipMemsetAsync(ws + 4096, 0, zero_bytes, stream);

    const int gNE = (n * HD + 255) / 256;    // node*feature grids
    const int gEE = (e * HD + 255) / 256;    // edge*feature grid

    // 1) scalar stats + degree/one-hot stats + features
    k_reduce_est<<<(n + 255) / 256, 256, 0, stream>>>(est, sums, n);
    k_edge_stats<<<(e + 255) / 256, 256, 0, stream>>>(edges, status, in_deg, out_deg,
                                                      in_stats, out_stats, e);
    k_node_feats<<<(n + 255) / 256, 256, 0, stream>>>(status, est, in_deg, out_deg,
                                                      in_stats, out_stats, sums,
                                                      feats, dinv, n1, n);

    // 2) GCN conv 1: h = feats@W1 ; agg ; +b1 ; relu         (bufA -> bufB)
    k_gemm_fin<<<gNE, 256, 0, stream>>>(feats, W1, bufA, n);
    k_selfloop<<<gNE, 256, 0, stream>>>(bufA, dinv, bufB, n);
    k_edge_agg<<<gEE, 256, 0, stream>>>(edges, bufA, dinv, bufB, e);
    k_bias_relu<<<gNE, 256, 0, stream>>>(bufB, b1, n);

    // 3) GCN conv 2: h = x1@W2 (WMMA) ; agg ; +b2 ; relu     (bufB -> bufA -> bufB)
    k_gemm_wmma<<<512, 256, HD * HD * 4, stream>>>(bufB, W2, nullptr, 0, bufA, n);
    k_selfloop<<<gNE, 256, 0, stream>>>(bufA, dinv, bufB, n);
    k_edge_agg<<<gEE, 256, 0, stream>>>(edges, bufA, dinv, bufB, e);
    k_bias_relu<<<gNE, 256, 0, stream>>>(bufB, b2, n);

    // 4) global mean + constant half of actor MLP
    k_colsum<<<(n + 511) / 512, 128, 0, stream>>>(bufB, gsum, n);
    k_gterm<<<1, 128, 0, stream>>>(gsum, Wa1, ba1, gmean, gterm, n);

    // 5) actor: hA = relu(x2@Wa1_top + gterm) (WMMA) ; logits = hA.Wa2 + mask
    k_gemm_wmma<<<512, 256, HD * HD * 4, stream>>>(bufB, Wa1, gterm, 1, bufA, n);
    k_logits<<<(n + 7) / 8, 256, 0, stream>>>(bufA, Wa2, ba2, status, n1, logits, n);

    // 6) value head
    k_value<<<1, 256, 0, stream>>>(gmean, Wc1, bc1, Wc2, bc2, Wc3, bc3, value);
}